// GuidedFilter_6528350290423
// MI455X (gfx1250) — compile-verified
//
#include <hip/hip_runtime.h>

// Guided filter, 48 planes of 512x512 fp32, radius 8 (17x17 box).
// Separable box filters as banded-ones matmuls on the fp32 WMMA pipe
// (V_WMMA_F32_16X16X4_F32). Zero-padding is folded into the constant band
// operands (column validity -> Sh/B, row validity -> Sv/A): data loads are
// unconditional clamped loads, no divergent branches. All global accesses
// use unsigned 32-bit element indices off uniform plane bases so the
// compiler emits SADDR + VADDR32 (scale_offset) addressing.

#define HH  512
#define WW  512
#define NPL 48
#define RAD 8
#define PITCH 146   // even -> 8B-aligned v2f LDS rows

typedef __attribute__((ext_vector_type(2))) float v2f;
typedef __attribute__((ext_vector_type(4))) float v4f;
typedef __attribute__((ext_vector_type(8))) float v8f;

__device__ __forceinline__ v8f wmma4(v2f a, v2f b, v8f c) {
  // D = A(16x4,f32) * B(4x16,f32) + C(16x16,f32), full fp32 precision
  return __builtin_amdgcn_wmma_f32_16x16x4_f32(
      /*neg_a=*/false, a, /*neg_b=*/false, b,
      /*c_mod=*/(short)0, c, /*reuse_a=*/false, /*reuse_b=*/false);
}

// band indicator: 1 iff m <= c <= m+16 (17-wide window), times validity mask
__device__ __forceinline__ float band(int c, int m, bool ok) {
  return (ok && c >= m && c <= m + 16) ? 1.0f : 0.0f;
}

__device__ __forceinline__ int clampi(int v, int lo, int hi) {
  return min(max(v, lo), hi);
}

// ---------------- Kernel 1: box(I), box(p), box(I*p), box(I*I) -> {a,b} ----

__global__ __launch_bounds__(256) void gf_stats(const float* __restrict__ I,
                                                const float* __restrict__ P,
                                                float* __restrict__ wsAB) {
  // Row-sum buffer, transposed: [quantity][col 0..15][local row 0..143].
  __shared__ float hb[4][16][PITCH];

  const int tid  = threadIdx.x;
  const int lane = tid & 31;
  const int wave = tid >> 5;
  const int half = lane >> 4;   // 0: lanes 0-15, 1: lanes 16-31
  const int m16  = lane & 15;   // A-row / B-col index within the 16-dim

  const int x0      = blockIdx.x * 16;   // output col tile
  const int rowBase = blockIdx.y * 128;  // first output row of this block
  // uniform (SGPR) plane bases, vector-typed for 32-bit scaled indexing
  const v2f* I2 = (const v2f*)(I    + (size_t)blockIdx.z * (HH * WW));
  const v2f* P2 = (const v2f*)(P    + (size_t)blockIdx.z * (HH * WW));
  v2f*       W2 = (v2f*)      (wsAB + (size_t)blockIdx.z * (2 * HH * WW));

  // Sh band * column validity: invariant over the tile loop -> hoist
  v2f bS[8];
  unsigned xoff2[8];                       // clamped col offset, v2f units
#pragma unroll
  for (int k = 0; k < 8; ++k) {
    const int c0  = 4 * k + 2 * half;
    const int gx0 = x0 - 8 + c0;
    bS[k] = {band(c0,     m16, (unsigned)gx0       < (unsigned)WW),
             band(c0 + 1, m16, (unsigned)(gx0 + 1) < (unsigned)WW)};
    xoff2[k] = (unsigned)clampi(gx0, 0, WW - 2) >> 1;
  }

  // ---------- H phase: row sums for local rows [0,144) ----------
  for (int t = wave; t < 9; t += 8) {
    const int grow = rowBase - 8 + t * 16 + m16;               // A-row of lane
    const unsigned row2 = (unsigned)clampi(grow, 0, HH - 1) * (WW / 2);

    v2f li[8], lp[8];
#pragma unroll
    for (int k = 0; k < 8; ++k) {          // unconditional b64 loads, GVS form
      li[k] = I2[row2 + xoff2[k]];
      lp[k] = P2[row2 + xoff2[k]];
    }

    v8f accI = {}, accP = {}, accIP = {}, accII = {};
#pragma unroll
    for (int k = 0; k < 8; ++k) {
      const v2f aI  = li[k];
      const v2f aP  = lp[k];
      const v2f aIP = {aI.x * aP.x, aI.y * aP.y};
      const v2f aII = {aI.x * aI.x, aI.y * aI.y};
      accI  = wmma4(aI,  bS[k], accI);
      accP  = wmma4(aP,  bS[k], accP);
      accIP = wmma4(aIP, bS[k], accIP);
      accII = wmma4(aII, bS[k], accII);
    }
    // D layout -> transposed LDS: vgpr i holds rows i (lanes<16) / i+8 (>=16)
#pragma unroll
    for (int i = 0; i < 8; ++i) {
      const int lr = t * 16 + i + 8 * half;
      hb[0][m16][lr] = accI[i];
      hb[1][m16][lr] = accP[i];
      hb[2][m16][lr] = accIP[i];
      hb[3][m16][lr] = accII[i];
    }
  }
  __syncthreads();

  // ---------- V phase: wave w owns output rows y0..y0+15 ----------
  const int y0 = rowBase + wave * 16;
  v2f bI[8], bP[8], bIP[8], bII[8];
#pragma unroll
  for (int k = 0; k < 8; ++k) {                        // ds_load_b64 burst
    const int lr = wave * 16 + 4 * k + 2 * half;       // even -> 8B aligned
    bI[k]  = *(const v2f*)&hb[0][m16][lr];
    bP[k]  = *(const v2f*)&hb[1][m16][lr];
    bIP[k] = *(const v2f*)&hb[2][m16][lr];
    bII[k] = *(const v2f*)&hb[3][m16][lr];
  }
  v8f sI = {}, sP = {}, sIP = {}, sII = {};
#pragma unroll
  for (int k = 0; k < 8; ++k) {
    const int c0 = 4 * k + 2 * half;
    const int gr = y0 - 8 + c0;                        // global H row
    const v2f aS = {band(c0,     m16, (unsigned)gr       < (unsigned)HH),
                    band(c0 + 1, m16, (unsigned)(gr + 1) < (unsigned)HH)};
    sI  = wmma4(aS, bI[k],  sI);
    sP  = wmma4(aS, bP[k],  sP);
    sIP = wmma4(aS, bIP[k], sIP);
    sII = wmma4(aS, bII[k], sII);
  }

  // ---------- epilogue: analytic N, compute a,b, b64 store interleaved ------
  const int x = x0 + m16;
  const float cx = (float)(min(x + RAD, WW - 1) - max(x - RAD, 0) + 1);
#pragma unroll
  for (int i = 0; i < 8; ++i) {
    const int y = y0 + i + 8 * half;   // D layout: vgpr i -> row i / i+8
    const float cy = (float)(min(y + RAD, HH - 1) - max(y - RAD, 0) + 1);
    const float invN = 1.0f / (cy * cx);
    const float mI  = sI[i]  * invN;
    const float mP  = sP[i]  * invN;
    const float mIP = sIP[i] * invN;
    const float mII = sII[i] * invN;
    const float cov = mIP - mI * mP;
    const float var = mII - mI * mI;
    const float a   = cov / (var + 0.01f);
    const float b   = mP - a * mI;
    W2[(unsigned)(y * WW + x)] = (v2f){a, b};          // b64 store, GVS form
  }
}

// ---------------- Kernel 2: box(a), box(b) -> q = mean_a * I + mean_b ------

__global__ __launch_bounds__(256) void gf_apply(const float* __restrict__ wsAB,
                                                const float* __restrict__ I,
                                                float* __restrict__ out) {
  __shared__ float hb[2][16][PITCH];

  const int tid  = threadIdx.x;
  const int lane = tid & 31;
  const int wave = tid >> 5;
  const int half = lane >> 4;
  const int m16  = lane & 15;

  const int x0      = blockIdx.x * 16;
  const int rowBase = blockIdx.y * 128;
  const v4f*   W4   = (const v4f*)(wsAB + (size_t)blockIdx.z * (2 * HH * WW));
  const float* Ipl  = I   + (size_t)blockIdx.z * (HH * WW);
  float*       outpl = out + (size_t)blockIdx.z * (HH * WW);

  v2f bS[8];
  unsigned xoff4[8];                       // clamped col offset, v4f units
#pragma unroll
  for (int k = 0; k < 8; ++k) {
    const int c0  = 4 * k + 2 * half;
    const int gx0 = x0 - 8 + c0;
    bS[k] = {band(c0,     m16, (unsigned)gx0       < (unsigned)WW),
             band(c0 + 1, m16, (unsigned)(gx0 + 1) < (unsigned)WW)};
    xoff4[k] = (unsigned)clampi(gx0, 0, WW - 2) >> 1;  // {a,b} pair index
  }

  for (int t = wave; t < 9; t += 8) {
    const int grow = rowBase - 8 + t * 16 + m16;
    const unsigned row4 = (unsigned)clampi(grow, 0, HH - 1) * (WW / 2);

    v4f lab[8];
#pragma unroll
    for (int k = 0; k < 8; ++k)            // one b128 load per chunk, GVS form
      lab[k] = W4[row4 + xoff4[k]];        // {a0,b0,a1,b1}, 16B aligned

    v8f accA = {}, accB = {};
#pragma unroll
    for (int k = 0; k < 8; ++k) {
      const v2f aA = {lab[k].x, lab[k].z};
      const v2f aB = {lab[k].y, lab[k].w};
      accA = wmma4(aA, bS[k], accA);
      accB = wmma4(aB, bS[k], accB);
    }
#pragma unroll
    for (int i = 0; i < 8; ++i) {
      const int lr = t * 16 + i + 8 * half;
      hb[0][m16][lr] = accA[i];
      hb[1][m16][lr] = accB[i];
    }
  }
  __syncthreads();

  const int y0 = rowBase + wave * 16;
  v2f ba[8], bb[8];
#pragma unroll
  for (int k = 0; k < 8; ++k) {
    const int lr = wave * 16 + 4 * k + 2 * half;
    ba[k] = *(const v2f*)&hb[0][m16][lr];
    bb[k] = *(const v2f*)&hb[1][m16][lr];
  }
  v8f sA = {}, sB = {};
#pragma unroll
  for (int k = 0; k < 8; ++k) {
    const int c0 = 4 * k + 2 * half;
    const int gr = y0 - 8 + c0;
    const v2f aS = {band(c0,     m16, (unsigned)gr       < (unsigned)HH),
                    band(c0 + 1, m16, (unsigned)(gr + 1) < (unsigned)HH)};
    sA = wmma4(aS, ba[k], sA);
    sB = wmma4(aS, bb[k], sB);
  }

  const int x = x0 + m16;
  const float cx = (float)(min(x + RAD, WW - 1) - max(x - RAD, 0) + 1);
#pragma unroll
  for (int i = 0; i < 8; ++i) {
    const int y = y0 + i + 8 * half;
    const float cy = (float)(min(y + RAD, HH - 1) - max(y - RAD, 0) + 1);
    const float invN = 1.0f / (cy * cx);
    const unsigned off = (unsigned)(y * WW + x);
    outpl[off] = (sA[i] * invN) * Ipl[off] + (sB[i] * invN);
  }
}

// ---------------- launch ----------------

extern "C" void kernel_launch(void* const* d_in, const int* in_sizes, int n_in,
                              void* d_out, int out_size, void* d_ws, size_t ws_size,
                              hipStream_t stream) {
  const float* I = (const float*)d_in[0];
  const float* P = (const float*)d_in[1];
  float* out = (float*)d_out;
  float* wsAB = (float*)d_ws;                      // 48 * 512*512 * {a,b} pairs

  dim3 grid(WW / 16, HH / 128, NPL);               // 32 x 4 x 48 workgroups
  dim3 block(256);                                  // 8 wave32 per WG

  gf_stats<<<grid, block, 0, stream>>>(I, P, wsAB);
  gf_apply<<<grid, block, 0, stream>>>(wsAB, I, out);
}